// FluxGNN_49357764165713
// MI455X (gfx1250) — compile-verified
//
#include <hip/hip_runtime.h>
#include <math.h>

typedef float v2f __attribute__((ext_vector_type(2)));
typedef float v8f __attribute__((ext_vector_type(8)));

__device__ __forceinline__ float gelu_exact(float x) {
    // exact GELU: 0.5*x*(1+erf(x/sqrt(2)))
    return 0.5f * x * (1.0f + erff(x * 0.70710678118654752440f));
}

// ---------------------------------------------------------------------------
// Kernel 1: per-node embedding h = mlp(u; pn) + u @ ip_W + ip_b   (N x 64)
// 64 threads per node, 4 nodes per 256-thread block, hidden layer in LDS.
// ---------------------------------------------------------------------------
__global__ __launch_bounds__(256) void node_h_kernel(
    const float* __restrict__ node_u,
    const float* __restrict__ pn_W1, const float* __restrict__ pn_b1,
    const float* __restrict__ pn_W2, const float* __restrict__ pn_b2,
    const float* __restrict__ ip_W,  const float* __restrict__ ip_b,
    float* __restrict__ h, int N)
{
    __shared__ float hid[4][64];
    const int sub  = threadIdx.x >> 6;
    const int j    = threadIdx.x & 63;
    const int node = blockIdx.x * 4 + sub;
    const bool ok  = node < N;

    float u0 = 0.f, u1 = 0.f, u2 = 0.f, u3 = 0.f, u4 = 0.f;
    if (ok) {
        const float* up = node_u + (size_t)node * 5;
        u0 = up[0]; u1 = up[1]; u2 = up[2]; u3 = up[3]; u4 = up[4];
    }
    float a = pn_b1[j]
            + u0 * pn_W1[0 * 64 + j] + u1 * pn_W1[1 * 64 + j]
            + u2 * pn_W1[2 * 64 + j] + u3 * pn_W1[3 * 64 + j]
            + u4 * pn_W1[4 * 64 + j];
    hid[sub][j] = gelu_exact(a);
    __syncthreads();

    float out = pn_b2[j] + ip_b[j]
              + u0 * ip_W[0 * 64 + j] + u1 * ip_W[1 * 64 + j]
              + u2 * ip_W[2 * 64 + j] + u3 * ip_W[3 * 64 + j]
              + u4 * ip_W[4 * 64 + j];
    #pragma unroll 8
    for (int k = 0; k < 64; ++k) out += hid[sub][k] * pn_W2[k * 64 + j];
    if (ok) h[(size_t)node * 64 + j] = out;
}

// ---------------------------------------------------------------------------
// Kernel 2: edge pipeline. One 256-thread block (8 wave32) per 16 edges.
// All GEMMs run on V_WMMA_F32_16X16X4_F32 (fp32 matrix pipe).
// LDS row strides are odd so lane-column reads of A hit distinct banks.
// ---------------------------------------------------------------------------
#define MSGP 297   // 292 msg channels, odd stride
#define HIDP 133   // 128 hidden channels, odd stride
#define MP   69    // 64 m channels, odd stride
#define HHP  229   // 224 head-hidden channels

__global__ __launch_bounds__(256) void edge_kernel(
    const float* __restrict__ node_u,
    const float* __restrict__ edge_attr,
    const float* __restrict__ mean_mom,
    const float* __restrict__ std_mom,
    const float* __restrict__ edge_memory,
    const int*   __restrict__ edge_index,
    const float* __restrict__ h,
    const float* __restrict__ pe_W1,  const float* __restrict__ pe_b1,
    const float* __restrict__ pe_W2,  const float* __restrict__ pe_b2,
    const float* __restrict__ pm_W1,  const float* __restrict__ pm_b1,
    const float* __restrict__ pm_W2,  const float* __restrict__ pm_b2,
    const float* __restrict__ prho_W1, const float* __restrict__ prho_b1,
    const float* __restrict__ prho_W2, const float* __restrict__ prho_b2,
    const float* __restrict__ pen_W1, const float* __restrict__ pen_b1,
    const float* __restrict__ pen_W2, const float* __restrict__ pen_b2,
    const float* __restrict__ pmu_W1, const float* __restrict__ pmu_b1,
    const float* __restrict__ pmu_W2, const float* __restrict__ pmu_b2,
    const float* __restrict__ pv_W1,  const float* __restrict__ pv_b1,
    const float* __restrict__ pv_W2,  const float* __restrict__ pv_b2,
    const float* __restrict__ ln_g,   const float* __restrict__ ln_b,
    float* __restrict__ out, int E)
{
    __shared__ float sMsg[16][MSGP];   // msg_in (292 cols)
    __shared__ float sHid[16][HIDP];   // gelu(msg @ pm_W1 + b1) (128 cols)
    __shared__ float sM[16][MP];       // m (64 cols), normalized in place
    __shared__ float sHH[16][HHP];     // head hidden: rho(64)|en(64)|mu(64)|v(32)
    __shared__ float sEH[16][32];      // eps hidden
    __shared__ float sGeo[16][16];     // nx,ny,tx,ty,r,vp0..3,du0,du1,du3,du4
    __shared__ float sOut[16][8];      // a_rho,a_e,amu0,amu1,alpha
    __shared__ float sStat[16][2];     // mean, rstd
    __shared__ int   sSrc[16], sDst[16];

    const int tid  = threadIdx.x;
    const int lane = tid & 31;
    const int wave = tid >> 5;
    const int e0   = blockIdx.x * 16;

    // ---- phase 0: per-edge geometry + velocity projections ----
    if (tid < 16) {
        const int ge = e0 + tid;
        const int s  = edge_index[ge];
        const int d  = edge_index[E + ge];
        sSrc[tid] = s; sDst[tid] = d;
        const float dx = edge_attr[(size_t)ge * 3 + 0];
        const float dy = edge_attr[(size_t)ge * 3 + 1];
        const float r  = edge_attr[(size_t)ge * 3 + 2];
        const float inv = 1.0f / (r + 1e-12f);
        const float nx = dx * inv, ny = dy * inv;
        const float tx = -ny, ty = nx;
        const float s0 = std_mom[0], s1 = std_mom[1];
        const float m0 = mean_mom[0], m1 = mean_mom[1];
        const float* us = node_u + (size_t)s * 5;
        const float* ud = node_u + (size_t)d * 5;
        const float rsx = us[3] * s0 + m0, rsy = us[4] * s1 + m1;
        const float rdx = ud[3] * s0 + m0, rdy = ud[4] * s1 + m1;
        const float iscale = 1.0f / (sqrtf(s0 * s0 + s1 * s1) + 1e-8f);
        sGeo[tid][0] = nx; sGeo[tid][1] = ny;
        sGeo[tid][2] = tx; sGeo[tid][3] = ty;
        sGeo[tid][4] = r;
        sGeo[tid][5] = (rsx * nx + rsy * ny) * iscale;
        sGeo[tid][6] = (rsx * tx + rsy * ty) * iscale;
        sGeo[tid][7] = (rdx * nx + rdy * ny) * iscale;
        sGeo[tid][8] = (rdx * tx + rdy * ty) * iscale;
        sGeo[tid][9]  = ud[0] - us[0];
        sGeo[tid][10] = ud[1] - us[1];
        sGeo[tid][11] = ud[3] - us[3];
        sGeo[tid][12] = ud[4] - us[4];
    }
    __syncthreads();

    // ---- eps hidden (1->32) and h gathers (build msg_in cols 0..255) ----
    #pragma unroll
    for (int it = 0; it < 2; ++it) {
        const int idx = tid + it * 256;          // 512 jobs
        const int e = idx >> 5, jj = idx & 31;
        const float r = sGeo[e][4];
        sEH[e][jj] = gelu_exact(r * pe_W1[jj] + pe_b1[jj]);
    }
    #pragma unroll
    for (int it = 0; it < 4; ++it) {
        const int idx = tid + it * 256;          // 1024 jobs
        const int e = idx >> 6, c = idx & 63;
        const float hs = h[(size_t)sSrc[e] * 64 + c];
        const float hd = h[(size_t)sDst[e] * 64 + c];
        sMsg[e][c]       = hs;
        sMsg[e][64 + c]  = hd;
        sMsg[e][128 + c] = hs + hd;
        const float dd = hs - hd;
        sMsg[e][192 + c] = dd * dd;
    }
    __syncthreads();

    // ---- eps output (32->32) and vel_proj into msg_in cols 256..291 ----
    #pragma unroll
    for (int it = 0; it < 2; ++it) {
        const int idx = tid + it * 256;
        const int e = idx >> 5, jj = idx & 31;
        float acc = pe_b2[jj];
        #pragma unroll 8
        for (int k = 0; k < 32; ++k) acc += sEH[e][k] * pe_W2[k * 32 + jj];
        sMsg[e][256 + jj] = acc;
    }
    if (tid < 64) {
        const int e = tid >> 2, q = tid & 3;
        sMsg[e][288 + q] = sGeo[e][5 + q];
    }
    __syncthreads();

    // ---- GEMM1: (16x292) @ (292x128), one 16-col tile per wave ----
    {
        const int j = wave * 16 + (lane & 15);
        const int m = lane & 15;
        const int koff = (lane < 16) ? 0 : 2;     // ISA 16x4 f32 A/B layout
        v8f c = {};
        for (int kk = 0; kk < 73; ++kk) {
            const int k = kk * 4 + koff;
            v2f a, b;
            a.x = sMsg[m][k];            a.y = sMsg[m][k + 1];
            b.x = pm_W1[k * 128 + j];    b.y = pm_W1[(k + 1) * 128 + j];
            c = __builtin_amdgcn_wmma_f32_16x16x4_f32(
                    false, a, false, b, (short)0, c, false, false);
        }
        const float bias = pm_b1[j];
        const int rbase = (lane < 16) ? 0 : 8;
        #pragma unroll
        for (int i = 0; i < 8; ++i)
            sHid[rbase + i][j] = gelu_exact(c[i] + bias);
    }
    __syncthreads();

    // ---- GEMM2: (16x128) @ (128x64) + b + edge_memory; waves 0..3 ----
    if (wave < 4) {
        const int j = wave * 16 + (lane & 15);
        const int m = lane & 15;
        const int koff = (lane < 16) ? 0 : 2;
        v8f c = {};
        for (int kk = 0; kk < 32; ++kk) {
            const int k = kk * 4 + koff;
            v2f a, b;
            a.x = sHid[m][k];           a.y = sHid[m][k + 1];
            b.x = pm_W2[k * 64 + j];    b.y = pm_W2[(k + 1) * 64 + j];
            c = __builtin_amdgcn_wmma_f32_16x16x4_f32(
                    false, a, false, b, (short)0, c, false, false);
        }
        const float bias = pm_b2[j];
        const int rbase = (lane < 16) ? 0 : 8;
        #pragma unroll
        for (int i = 0; i < 8; ++i) {
            const int row = rbase + i;
            sM[row][j] = c[i] + bias + edge_memory[(size_t)(e0 + row) * 64 + j];
        }
    }
    __syncthreads();

    // ---- layernorm over 64 channels, write m output ----
    if (tid < 16) {
        float s = 0.f, ss = 0.f;
        #pragma unroll 8
        for (int c2 = 0; c2 < 64; ++c2) { const float v = sM[tid][c2]; s += v; ss += v * v; }
        const float mean = s * (1.0f / 64.0f);
        const float var  = ss * (1.0f / 64.0f) - mean * mean;
        sStat[tid][0] = mean;
        sStat[tid][1] = rsqrtf(var + 1e-5f);
    }
    __syncthreads();
    #pragma unroll
    for (int it = 0; it < 4; ++it) {
        const int idx = tid + it * 256;
        const int e = idx >> 6, c2 = idx & 63;
        const float v = (sM[e][c2] - sStat[e][0]) * sStat[e][1] * ln_g[c2] + ln_b[c2];
        sM[e][c2] = v;
        out[(size_t)10 * E + (size_t)(e0 + e) * 64 + c2] = v;   // m output
    }
    __syncthreads();

    // ---- head first layers: 14 concatenated 16-col WMMA tiles ----
    for (int tt = wave; tt < 14; tt += 8) {
        const float* Wt; const float* bt; int wcols, cw, dcol;
        if (tt < 4)       { Wt = prho_W1; bt = prho_b1; wcols = 64; cw = tt * 16;        dcol = tt * 16; }
        else if (tt < 8)  { Wt = pen_W1;  bt = pen_b1;  wcols = 64; cw = (tt - 4) * 16;  dcol = 64 + (tt - 4) * 16; }
        else if (tt < 12) { Wt = pmu_W1;  bt = pmu_b1;  wcols = 64; cw = (tt - 8) * 16;  dcol = 128 + (tt - 8) * 16; }
        else              { Wt = pv_W1;   bt = pv_b1;   wcols = 32; cw = (tt - 12) * 16; dcol = 192 + (tt - 12) * 16; }
        const int j = cw + (lane & 15);
        const int m = lane & 15;
        const int koff = (lane < 16) ? 0 : 2;
        v8f c = {};
        for (int kk = 0; kk < 16; ++kk) {
            const int k = kk * 4 + koff;
            v2f a, b;
            a.x = sM[m][k];           a.y = sM[m][k + 1];
            b.x = Wt[k * wcols + j];  b.y = Wt[(k + 1) * wcols + j];
            c = __builtin_amdgcn_wmma_f32_16x16x4_f32(
                    false, a, false, b, (short)0, c, false, false);
        }
        const float bias = bt[j];
        const int rbase = (lane < 16) ? 0 : 8;
        const int dj = dcol + (lane & 15);
        #pragma unroll
        for (int i = 0; i < 8; ++i)
            sHH[rbase + i][dj] = gelu_exact(c[i] + bias);
    }
    __syncthreads();

    // ---- head second layers (5 scalar outputs per edge) ----
    if (tid < 80) {
        const int e = tid & 15, o = tid >> 4;
        float acc;
        if (o == 0) {
            acc = prho_b2[0];
            #pragma unroll 8
            for (int k = 0; k < 64; ++k) acc += sHH[e][k] * prho_W2[k];
            sOut[e][0] = acc;
        } else if (o == 1) {
            acc = pen_b2[0];
            #pragma unroll 8
            for (int k = 0; k < 64; ++k) acc += sHH[e][64 + k] * pen_W2[k];
            sOut[e][1] = acc;
        } else if (o == 2) {
            acc = pmu_b2[0];
            #pragma unroll 8
            for (int k = 0; k < 64; ++k) acc += sHH[e][128 + k] * pmu_W2[2 * k + 0];
            sOut[e][2] = acc;
        } else if (o == 3) {
            acc = pmu_b2[1];
            #pragma unroll 8
            for (int k = 0; k < 64; ++k) acc += sHH[e][128 + k] * pmu_W2[2 * k + 1];
            sOut[e][3] = acc;
        } else {
            acc = pv_b2[0];
            #pragma unroll 8
            for (int k = 0; k < 32; ++k) acc += sHH[e][192 + k] * pv_W2[k];
            sOut[e][4] = 1.0f / (1.0f + expf(-acc));   // sigmoid -> alpha
        }
    }
    __syncthreads();

    // ---- flux assembly + remaining outputs ----
    if (tid < 16) {
        const int e = tid, ge = e0 + e;
        const float nx = sGeo[e][0], ny = sGeo[e][1];
        const float tx = sGeo[e][2], ty = sGeo[e][3];
        const float r  = sGeo[e][4];
        const float du0 = sGeo[e][9], du1 = sGeo[e][10];
        const float du3 = sGeo[e][11], du4 = sGeo[e][12];
        const float a_rho = sOut[e][0], a_e = sOut[e][1];
        const float am0 = sOut[e][2], am1 = sOut[e][3], al = sOut[e][4];

        const float c0 = a_rho - al * du0;              // F_rho
        out[(size_t)ge * 2 + 0] = c0 * nx;
        out[(size_t)ge * 2 + 1] = c0 * ny;
        const float c1 = a_e - al * du1;                // F_e
        out[(size_t)2 * E + (size_t)ge * 2 + 0] = c1 * nx;
        out[(size_t)2 * E + (size_t)ge * 2 + 1] = c1 * ny;
        out[(size_t)4 * E + (size_t)ge * 2 + 0] = am0 * nx + am1 * tx - al * du3;  // F_rhou
        out[(size_t)4 * E + (size_t)ge * 2 + 1] = am0 * ny + am1 * ty - al * du4;
        out[(size_t)6 * E + (size_t)ge * 2 + 0] = nx;   // n
        out[(size_t)6 * E + (size_t)ge * 2 + 1] = ny;
        out[(size_t)8 * E + ge] = r;                    // r
        out[(size_t)9 * E + ge] = al;                   // alpha
    }
}

// ---------------------------------------------------------------------------
extern "C" void kernel_launch(void* const* d_in, const int* in_sizes, int n_in,
                              void* d_out, int out_size, void* d_ws, size_t ws_size,
                              hipStream_t stream) {
    const float* node_u    = (const float*)d_in[0];
    const float* edge_attr = (const float*)d_in[1];
    const float* mean_mom  = (const float*)d_in[2];
    const float* std_mom   = (const float*)d_in[3];
    const float* edge_mem  = (const float*)d_in[4];
    const int*   edge_idx  = (const int*)d_in[5];
    const float* pn_W1 = (const float*)d_in[6];
    const float* pn_b1 = (const float*)d_in[7];
    const float* pn_W2 = (const float*)d_in[8];
    const float* pn_b2 = (const float*)d_in[9];
    const float* ip_W  = (const float*)d_in[10];
    const float* ip_b  = (const float*)d_in[11];
    const float* pe_W1 = (const float*)d_in[12];
    const float* pe_b1 = (const float*)d_in[13];
    const float* pe_W2 = (const float*)d_in[14];
    const float* pe_b2 = (const float*)d_in[15];
    const float* pm_W1 = (const float*)d_in[16];
    const float* pm_b1 = (const float*)d_in[17];
    const float* pm_W2 = (const float*)d_in[18];
    const float* pm_b2 = (const float*)d_in[19];
    const float* prho_W1 = (const float*)d_in[20];
    const float* prho_b1 = (const float*)d_in[21];
    const float* prho_W2 = (const float*)d_in[22];
    const float* prho_b2 = (const float*)d_in[23];
    const float* pen_W1 = (const float*)d_in[24];
    const float* pen_b1 = (const float*)d_in[25];
    const float* pen_W2 = (const float*)d_in[26];
    const float* pen_b2 = (const float*)d_in[27];
    const float* pmu_W1 = (const float*)d_in[28];
    const float* pmu_b1 = (const float*)d_in[29];
    const float* pmu_W2 = (const float*)d_in[30];
    const float* pmu_b2 = (const float*)d_in[31];
    const float* pv_W1 = (const float*)d_in[32];
    const float* pv_b1 = (const float*)d_in[33];
    const float* pv_W2 = (const float*)d_in[34];
    const float* pv_b2 = (const float*)d_in[35];
    const float* ln_g  = (const float*)d_in[36];
    const float* ln_b  = (const float*)d_in[37];

    const int N = in_sizes[0] / 5;
    const int E = in_sizes[1] / 3;
    float* h = (float*)d_ws;   // N x 64 staging for node embeddings

    node_h_kernel<<<(N + 3) / 4, 256, 0, stream>>>(
        node_u, pn_W1, pn_b1, pn_W2, pn_b2, ip_W, ip_b, h, N);

    // E is a multiple of 16 (500000 / 16 = 31250): full blocks, EXEC all-ones.
    edge_kernel<<<E / 16, 256, 0, stream>>>(
        node_u, edge_attr, mean_mom, std_mom, edge_mem, edge_idx, h,
        pe_W1, pe_b1, pe_W2, pe_b2,
        pm_W1, pm_b1, pm_W2, pm_b2,
        prho_W1, prho_b1, prho_W2, prho_b2,
        pen_W1, pen_b1, pen_W2, pen_b2,
        pmu_W1, pmu_b1, pmu_W2, pmu_b2,
        pv_W1, pv_b1, pv_W2, pv_b2,
        ln_g, ln_b,
        (float*)d_out, E);
}